// UnifiedMambaBlock_60610578482001
// MI455X (gfx1250) — compile-verified
//
#include <hip/hip_runtime.h>
#include <hip/hip_bf16.h>
#include <math.h>

// Problem constants (match reference)
#define BATCH   2
#define LSEQ    2048
#define DMODEL  768
#define DINNER  1536
#define DSTATE  16
#define DTRANK  48
#define DCONV   4
#define MROWS   (BATCH * LSEQ)        // 4096 tokens
#define XDBL_W  (DTRANK + 2 * DSTATE) // 80

typedef __attribute__((ext_vector_type(2))) float v2f;
typedef __attribute__((ext_vector_type(8))) float v8f;

__device__ __forceinline__ float silu_f(float x) { return x / (1.0f + expf(-x)); }

#define WMMA_F32(A_, B_, C_) \
    __builtin_amdgcn_wmma_f32_16x16x4_f32(false, (A_), false, (B_), (short)0, (C_), false, false)

#if defined(__has_builtin)
#if __has_builtin(__builtin_amdgcn_sched_group_barrier)
#define SCHED_GROUP(mask, n, id) __builtin_amdgcn_sched_group_barrier((mask), (n), (id))
#else
#define SCHED_GROUP(mask, n, id)
#endif
#else
#define SCHED_GROUP(mask, n, id)
#endif

// Issue the 8 WMMAs for one fragment set
#define MMA_SET(P)                            \
    acc00 = WMMA_F32(P##_a0, P##_b0, acc00);  \
    acc01 = WMMA_F32(P##_a0, P##_b1, acc01);  \
    acc02 = WMMA_F32(P##_a0, P##_b2, acc02);  \
    acc03 = WMMA_F32(P##_a0, P##_b3, acc03);  \
    acc10 = WMMA_F32(P##_a1, P##_b0, acc10);  \
    acc11 = WMMA_F32(P##_a1, P##_b1, acc11);  \
    acc12 = WMMA_F32(P##_a1, P##_b2, acc12);  \
    acc13 = WMMA_F32(P##_a1, P##_b3, acc13);

// ---------------------------------------------------------------------------
// fp32 WMMA GEMM: Out[M,N] = act(A[M,K] @ W[N,K]^T [+ bias])
//   A row-major (row stride lda); W row-major [N,K]; Out row stride ldo.
//   act==0: none; act==1: softplus(x + bias[col])
// Wave tile 32x64: 2 A-fragments x 4 B-fragments -> 8 wmma per 6 loads.
// Block = 8 waves, 4(M) x 2(N) -> 128x128 macro tile.
// 2-deep ping-pong pipeline, with sched_group_barrier enforcing the
// [6 loads | 8 wmma | 6 loads | 8 wmma] loop shape so each WMMA group
// waits with a full iteration of loadcnt slack instead of wait-all-0.
// Requires K % 8 == 0 (true for all call sites: 768 / 1536 / 48).
// ---------------------------------------------------------------------------
__global__ __launch_bounds__(256)
void gemm_wmma_f32(const float* __restrict__ A, int lda,
                   const float* __restrict__ W, int K,
                   float* __restrict__ Out, int ldo, int N,
                   const float* __restrict__ bias, int act)
{
    const int tid   = threadIdx.x;
    const int wave  = tid >> 5;
    const int lane  = tid & 31;
    const int mL    = lane & 15;          // row within A tile / col within B tile
    const int kh    = (lane >> 4) << 1;   // 0 or 2 : K sub-slot per half-wave
    const int waveM = wave >> 1;          // 0..3
    const int waveN = wave & 1;           // 0..1

    const int rowBase = blockIdx.y * 128 + waveM * 32;
    const int colBase = blockIdx.x * 128 + waveN * 64;

    // Clamp OOB columns into range (loaded garbage never stored).
    int n0 = colBase + 0 * 16 + mL; int c0 = (n0 < N) ? n0 : (N - 1);
    int n1 = colBase + 1 * 16 + mL; int c1 = (n1 < N) ? n1 : (N - 1);
    int n2 = colBase + 2 * 16 + mL; int c2 = (n2 < N) ? n2 : (N - 1);
    int n3 = colBase + 3 * 16 + mL; int c3 = (n3 < N) ? n3 : (N - 1);

    // Running pointers, advanced 8 floats per iteration
    const float* pA0 = A + (size_t)(rowBase + mL) * lda + kh;
    const float* pA1 = A + (size_t)(rowBase + 16 + mL) * lda + kh;
    const float* pW0 = W + (size_t)c0 * K + kh;
    const float* pW1 = W + (size_t)c1 * K + kh;
    const float* pW2 = W + (size_t)c2 * K + kh;
    const float* pW3 = W + (size_t)c3 * K + kh;

    v8f acc00 = {}, acc01 = {}, acc02 = {}, acc03 = {};
    v8f acc10 = {}, acc11 = {}, acc12 = {}, acc13 = {};

    // Two in-flight fragment sets (slices k and k+4)
    v2f f0_a0 = *(const v2f*)(pA0 + 0);
    v2f f0_a1 = *(const v2f*)(pA1 + 0);
    v2f f0_b0 = *(const v2f*)(pW0 + 0);
    v2f f0_b1 = *(const v2f*)(pW1 + 0);
    v2f f0_b2 = *(const v2f*)(pW2 + 0);
    v2f f0_b3 = *(const v2f*)(pW3 + 0);

    v2f f1_a0 = *(const v2f*)(pA0 + 4);
    v2f f1_a1 = *(const v2f*)(pA1 + 4);
    v2f f1_b0 = *(const v2f*)(pW0 + 4);
    v2f f1_b1 = *(const v2f*)(pW1 + 4);
    v2f f1_b2 = *(const v2f*)(pW2 + 4);
    v2f f1_b3 = *(const v2f*)(pW3 + 4);

    pA0 += 8; pA1 += 8; pW0 += 8; pW1 += 8; pW2 += 8; pW3 += 8;

#pragma unroll 1
    for (int k = 8; k + 8 <= K; k += 8) {
        // Prefetch slice k into fresh SSA values
        v2f t_a0 = *(const v2f*)(pA0 + 0);
        v2f t_a1 = *(const v2f*)(pA1 + 0);
        v2f t_b0 = *(const v2f*)(pW0 + 0);
        v2f t_b1 = *(const v2f*)(pW1 + 0);
        v2f t_b2 = *(const v2f*)(pW2 + 0);
        v2f t_b3 = *(const v2f*)(pW3 + 0);

        MMA_SET(f0)  // consume slice k-8

        f0_a0 = t_a0; f0_a1 = t_a1;
        f0_b0 = t_b0; f0_b1 = t_b1; f0_b2 = t_b2; f0_b3 = t_b3;

        // Prefetch slice k+4
        v2f u_a0 = *(const v2f*)(pA0 + 4);
        v2f u_a1 = *(const v2f*)(pA1 + 4);
        v2f u_b0 = *(const v2f*)(pW0 + 4);
        v2f u_b1 = *(const v2f*)(pW1 + 4);
        v2f u_b2 = *(const v2f*)(pW2 + 4);
        v2f u_b3 = *(const v2f*)(pW3 + 4);

        MMA_SET(f1)  // consume slice k-4

        f1_a0 = u_a0; f1_a1 = u_a1;
        f1_b0 = u_b0; f1_b1 = u_b1; f1_b2 = u_b2; f1_b3 = u_b3;

        pA0 += 8; pA1 += 8; pW0 += 8; pW1 += 8; pW2 += 8; pW3 += 8;

        // Enforce the pipelined loop shape: loads | wmma | loads | wmma
        SCHED_GROUP(0x020, 6, 0);  // 6 VMEM reads  (f0 prefetch)
        SCHED_GROUP(0x008, 8, 0);  // 8 WMMA        (consume f0)
        SCHED_GROUP(0x020, 6, 0);  // 6 VMEM reads  (f1 prefetch)
        SCHED_GROUP(0x008, 8, 0);  // 8 WMMA        (consume f1)
    }

    // Drain the last two slices
    MMA_SET(f0)
    MMA_SET(f1)

    // Store: VGPR i -> row (i + 8*(lane>>4)), col (lane&15) within tile.
    v8f acc[2][4] = {{acc00, acc01, acc02, acc03},
                     {acc10, acc11, acc12, acc13}};
    const int rOff = (lane >> 4) << 3;
#pragma unroll
    for (int r = 0; r < 2; ++r) {
#pragma unroll
        for (int j = 0; j < 4; ++j) {
            const int col = colBase + j * 16 + mL;
            if (col >= N) continue;
            const float bv = (act == 1) ? bias[col] : 0.0f;
#pragma unroll
            for (int i = 0; i < 8; ++i) {
                const int row = rowBase + r * 16 + rOff + i;
                float v = acc[r][j][i];
                if (act == 1) {
                    v += bv;
                    v = (v > 20.0f) ? v : log1pf(expf(v)); // softplus
                }
                Out[(size_t)row * ldo + col] = v;
            }
        }
    }
}

// ---------------------------------------------------------------------------
// Causal depthwise conv1d (taps=4) + bias + SiLU.  x = xz[:, 0:DINNER].
// ---------------------------------------------------------------------------
__global__ __launch_bounds__(256)
void conv_silu_kernel(const float* __restrict__ xz,
                      const float* __restrict__ w,    // [DINNER, 1, DCONV]
                      const float* __restrict__ bias, // [DINNER]
                      float* __restrict__ xc)         // [B,L,DINNER]
{
    const int g = blockIdx.x * blockDim.x + threadIdx.x;
    if (g >= BATCH * LSEQ * DINNER) return;
    const int d  = g % DINNER;
    const int t  = (g / DINNER) % LSEQ;
    const int bb = g / (DINNER * LSEQ);

    float acc = bias[d];
#pragma unroll
    for (int j = 0; j < DCONV; ++j) {
        const int tt = t - (DCONV - 1) + j;
        if (tt >= 0)
            acc += w[d * DCONV + j] *
                   xz[((size_t)(bb * LSEQ + tt)) * (2 * DINNER) + d];
    }
    xc[g] = silu_f(acc);
}

// ---------------------------------------------------------------------------
// Selective scan, one lane per (b, d, s).  16 lanes per channel -> shfl_xor
// tree reduction for y_t = sum_s h*C.  Fuses +D*x_conv and silu(z) gate.
// ---------------------------------------------------------------------------
__global__ __launch_bounds__(256)
void scan_kernel(const float* __restrict__ dt,    // [B,L,DINNER]
                 const float* __restrict__ xc,    // [B,L,DINNER]
                 const float* __restrict__ xdbl,  // [B,L,80] (B at 48, C at 64)
                 const float* __restrict__ xz,    // [B,L,2*DINNER] (z half)
                 const float* __restrict__ A_log, // [DINNER, DSTATE]
                 const float* __restrict__ Dp,    // [DINNER]
                 float* __restrict__ yg)          // [B,L,DINNER] gated output
{
    const int g  = blockIdx.x * blockDim.x + threadIdx.x; // < B*DINNER*DSTATE
    const int s  = g & (DSTATE - 1);
    const int d  = (g >> 4) % DINNER;
    const int bb = g / (DINNER * DSTATE);

    const float a  = -expf(A_log[d * DSTATE + s]);
    const float Dd = Dp[d];
    float h = 0.0f;

    for (int t = 0; t < LSEQ; ++t) {
        const size_t row = (size_t)bb * LSEQ + t;
        const float dtv = dt[row * DINNER + d];
        const float xv  = xc[row * DINNER + d];
        const float Bv  = xdbl[row * XDBL_W + DTRANK + s];
        const float Cv  = xdbl[row * XDBL_W + DTRANK + DSTATE + s];

        h = h * expf(dtv * a) + (xv * dtv) * Bv;
        float p = h * Cv;
        // reduce over the 16 states (xor masks < 16 stay inside the group)
        p += __shfl_xor(p, 1, 32);
        p += __shfl_xor(p, 2, 32);
        p += __shfl_xor(p, 4, 32);
        p += __shfl_xor(p, 8, 32);
        if (s == 0) {
            const float zv = xz[row * (2 * DINNER) + DINNER + d];
            yg[row * DINNER + d] = (p + Dd * xv) * silu_f(zv);
        }
    }
}

// ---------------------------------------------------------------------------
extern "C" void kernel_launch(void* const* d_in, const int* in_sizes, int n_in,
                              void* d_out, int out_size, void* d_ws, size_t ws_size,
                              hipStream_t stream)
{
    const float* u          = (const float*)d_in[0];
    const float* in_proj_w  = (const float*)d_in[1];
    const float* conv_w     = (const float*)d_in[2];
    const float* conv_b     = (const float*)d_in[3];
    const float* x_proj_w   = (const float*)d_in[4];
    const float* dt_proj_w  = (const float*)d_in[5];
    const float* dt_proj_b  = (const float*)d_in[6];
    const float* A_log      = (const float*)d_in[7];
    const float* Dp         = (const float*)d_in[8];
    const float* out_proj_w = (const float*)d_in[9];
    float* out = (float*)d_out;

    // Workspace carve-up (fp32): ~127 MB total
    float* xz   = (float*)d_ws;                      // [4096, 3072]
    float* xc   = xz   + (size_t)MROWS * 2 * DINNER; // [4096, 1536]
    float* xdbl = xc   + (size_t)MROWS * DINNER;     // [4096, 80]
    float* dtb  = xdbl + (size_t)MROWS * XDBL_W;     // [4096, 1536]
    float* yg   = dtb  + (size_t)MROWS * DINNER;     // [4096, 1536]

    const dim3 blk(256);

    // 1) in_proj: xz[4096,3072] = u @ in_proj_w^T   (K=768)
    gemm_wmma_f32<<<dim3((2 * DINNER) / 128, MROWS / 128), blk, 0, stream>>>(
        u, DMODEL, in_proj_w, DMODEL, xz, 2 * DINNER, 2 * DINNER, nullptr, 0);

    // 2) causal depthwise conv + bias + SiLU
    conv_silu_kernel<<<(BATCH * LSEQ * DINNER + 255) / 256, blk, 0, stream>>>(
        xz, conv_w, conv_b, xc);

    // 3) x_proj: xdbl[4096,80] = xc @ x_proj_w^T   (K=1536, edge N handled)
    gemm_wmma_f32<<<dim3((XDBL_W + 127) / 128, MROWS / 128), blk, 0, stream>>>(
        xc, DINNER, x_proj_w, DINNER, xdbl, XDBL_W, XDBL_W, nullptr, 0);

    // 4) dt: softplus(xdbl[:, :48] @ dt_proj_w^T + dt_proj_b)  (K=48)
    gemm_wmma_f32<<<dim3(DINNER / 128, MROWS / 128), blk, 0, stream>>>(
        xdbl, XDBL_W, dt_proj_w, DTRANK, dtb, DINNER, DINNER, dt_proj_b, 1);

    // 5) selective scan + skip + silu(z) gate
    scan_kernel<<<(BATCH * DINNER * DSTATE) / 256, blk, 0, stream>>>(
        dtb, xc, xdbl, xz, A_log, Dp, yg);

    // 6) out_proj: out[4096,768] = yg @ out_proj_w^T  (K=1536)
    gemm_wmma_f32<<<dim3(DMODEL / 128, MROWS / 128), blk, 0, stream>>>(
        yg, DINNER, out_proj_w, DINNER, out, DMODEL, DMODEL, nullptr, 0);
}